// ZBLRepulsion_65369402245345
// MI455X (gfx1250) — compile-verified
//
#include <hip/hip_runtime.h>
#include <hip/hip_bf16.h>
#include <stdint.h>

// ---------------------------------------------------------------------------
// ZBL repulsion for MI455X (gfx1250, wave32).
//   * Edge stream (24 B/edge * 8M = 192 MB) staged via CDNA5 async-to-LDS
//     (global_load_async_to_lds_b128, ASYNCcnt), tagged TH_LOAD_NT so the
//     once-read stream does not evict the hot L2 working set (1.6 MB atom
//     table + 0.8 MB atomic accumulator).  HBM roofline ~8.2 us.
//   * Per-wave self-consistency: each lane consumes only LDS bytes written by
//     its own wave's async loads, so s_wait_asynccnt 0 is sufficient — no
//     block barrier needed at all.
//   * Scatter via non-returning global_atomic_add_f32; reciprocals via
//     v_rcp_f32 (1 trans op instead of ~7-op IEEE divide expansions).
// ---------------------------------------------------------------------------

#define TILE  1024
#define BLOCK 256

// ---- CDNA5 async copy: 16 bytes global -> LDS per lane (ASYNCcnt) ---------
// Non-temporal: edge data is streamed exactly once.
__device__ __forceinline__ void async_cp_b128_nt(void* lds_dst, const void* gsrc) {
  // Low 32 bits of a flat LDS pointer are the 0-based LDS byte offset
  // (aperture mapping); hardware adds LDS_BASE.
  unsigned loff = (unsigned)(uintptr_t)lds_dst;
  unsigned long long g = (unsigned long long)(uintptr_t)gsrc;
  asm volatile("global_load_async_to_lds_b128 %0, %1, off th:TH_LOAD_NT"
               :: "v"(loff), "v"(g)
               : "memory");
}

__device__ __forceinline__ void wait_async0() {
  asm volatile("s_wait_asynccnt 0x0" ::: "memory");
}

// ---------------------------------------------------------------------------
struct ZblConsts {
  float acoef;              // |a_coefficient|
  float c0, c1, c2, c3;     // normalized |phi_coefficients|
  float e0, e1, e2, e3;     // -phi_exponents * log2(e)  (for v_exp_f32)
};

__device__ __forceinline__ void edge_compute(float draw, float swraw, float esraw,
                                             float bm, int ii, int jj,
                                             const float2* __restrict__ table,
                                             float* __restrict__ accum,
                                             const ZblConsts& K) {
  float d  = fmaxf(draw, 1e-6f);
  float sw = fmaxf(swraw, 0.0f);
  float es = fmaxf(esraw, 0.0f);

  float2 ti = table[ii];   // (za_i, safe_z_i)  -- L2-resident gather
  float2 tj = table[jj];

  float za_sum = fmaxf(ti.x + tj.x, 1e-10f) * bm;
  // a_ij = acoef / za_sum  (bm==0: rcp(0)=+inf -> arg=0 -> rep*bm=0, faithful)
  float a_ij = K.acoef * __builtin_amdgcn_rcpf(za_sum);
  float arg  = d * __builtin_amdgcn_rcpf(fmaxf(a_ij, 1e-10f));

  // phi = sum_k c_k * exp(-e_k * arg) == sum_k c_k * 2^(e'_k * arg)
  float phi = K.c0 * __builtin_amdgcn_exp2f(K.e0 * arg)
            + K.c1 * __builtin_amdgcn_exp2f(K.e1 * arg)
            + K.c2 * __builtin_amdgcn_exp2f(K.e2 * arg)
            + K.c3 * __builtin_amdgcn_exp2f(K.e3 * arg);

  float rep = (ti.y * tj.y * __builtin_amdgcn_rcpf(d) * phi + es) * sw * bm;
  atomicAdd(accum + ii, rep);                    // global_atomic_add_f32
}

// ---- Kernel 1: per-atom table (za, safe_z) + zero accumulator -------------
__global__ __launch_bounds__(256)
void zbl_init_kernel(const float* __restrict__ Z,
                     const float* __restrict__ a_exp_p,
                     float2* __restrict__ table,
                     float* __restrict__ accum, int N) {
  int a = blockIdx.x * blockDim.x + threadIdx.x;
  if (a < N) {
    float aexp = fabsf(a_exp_p[0]);
    float z    = fmaxf(Z[a], 1e-6f);
    // z^|aexp| = 2^(log2(z)*|aexp|)  via v_log_f32 / v_exp_f32
    float za   = __builtin_amdgcn_exp2f(__builtin_amdgcn_logf(z) * aexp);
    table[a]   = make_float2(za, z);
    accum[a]   = 0.0f;
  }
}

// ---- Kernel 2: edge stream through async-LDS tiles (barrier-free) ---------
__global__ __launch_bounds__(BLOCK)
void zbl_edges_kernel(const float* __restrict__ dist,
                      const float* __restrict__ swo,
                      const float* __restrict__ esh,
                      const float* __restrict__ bmask,
                      const int*   __restrict__ idx_i,
                      const int*   __restrict__ idx_j,
                      const float2* __restrict__ table,
                      const float* __restrict__ a_coef_p,
                      const float* __restrict__ phi_c,
                      const float* __restrict__ phi_e,
                      float* __restrict__ accum, int E) {
  __shared__ __align__(16) float s_d[TILE];
  __shared__ __align__(16) float s_sw[TILE];
  __shared__ __align__(16) float s_es[TILE];
  __shared__ __align__(16) float s_bm[TILE];
  __shared__ __align__(16) int   s_i[TILE];
  __shared__ __align__(16) int   s_j[TILE];

  const int tid = threadIdx.x;
  const int t4  = tid * 4;
  const long long base = (long long)blockIdx.x * TILE;

  // Uniform scalars (constant-cache s_loads)
  ZblConsts K;
  K.acoef = fabsf(a_coef_p[0]);
  float r0 = fabsf(phi_c[0]), r1 = fabsf(phi_c[1]);
  float r2 = fabsf(phi_c[2]), r3 = fabsf(phi_c[3]);
  float inv = 1.0f / fmaxf(r0 + r1 + r2 + r3, 1e-10f);
  K.c0 = r0 * inv; K.c1 = r1 * inv; K.c2 = r2 * inv; K.c3 = r3 * inv;
  const float L2E = 1.4426950408889634f;
  K.e0 = -phi_e[0] * L2E; K.e1 = -phi_e[1] * L2E;
  K.e2 = -phi_e[2] * L2E; K.e3 = -phi_e[3] * L2E;

  if (base + TILE <= (long long)E) {
    // Full tile: one NT async b128 per lane per array (perfectly coalesced;
    // no VGPRs held while in flight).  Each lane consumes only the bytes its
    // own wave wrote, so the per-wave asynccnt wait fully orders the data —
    // no block barrier required.
    async_cp_b128_nt(&s_d[t4],  dist  + base + t4);
    async_cp_b128_nt(&s_sw[t4], swo   + base + t4);
    async_cp_b128_nt(&s_es[t4], esh   + base + t4);
    async_cp_b128_nt(&s_bm[t4], bmask + base + t4);
    async_cp_b128_nt(&s_i[t4],  idx_i + base + t4);
    async_cp_b128_nt(&s_j[t4],  idx_j + base + t4);
    wait_async0();          // s_wait_asynccnt 0 (per-wave producer->consumer)
#pragma unroll
    for (int k = 0; k < 4; ++k) {
      int s = t4 + k;
      edge_compute(s_d[s], s_sw[s], s_es[s], s_bm[s], s_i[s], s_j[s],
                   table, accum, K);
    }
  } else {
    // Tail tile (fewer than TILE edges): guarded direct loads, no OOB async reads.
    for (int k = 0; k < 4; ++k) {
      long long e = base + t4 + k;
      if (e < (long long)E)
        edge_compute(dist[e], swo[e], esh[e], bmask[e], idx_i[e], idx_j[e],
                     table, accum, K);
    }
  }
}

// ---- Kernel 3: out = nan_to_num(accum * atom_mask) ------------------------
__global__ __launch_bounds__(256)
void zbl_final_kernel(float* __restrict__ out,
                      const float* __restrict__ amask, int N) {
  int a = blockIdx.x * blockDim.x + threadIdx.x;
  if (a < N) {
    float v = out[a] * amask[a];
    // nan/+inf/-inf -> 0  (fabs(NaN) <= X is false)
    out[a] = (fabsf(v) <= 3.402823466e+38f) ? v : 0.0f;
  }
}

// ---------------------------------------------------------------------------
extern "C" void kernel_launch(void* const* d_in, const int* in_sizes, int n_in,
                              void* d_out, int out_size, void* d_ws, size_t ws_size,
                              hipStream_t stream) {
  const float* Z     = (const float*)d_in[0];   // atomic_numbers [N]
  const float* dist  = (const float*)d_in[1];   // distances      [E]
  const float* swo   = (const float*)d_in[2];   // switch_off     [E]
  const float* esh   = (const float*)d_in[3];   // eshift         [E]
  const float* amask = (const float*)d_in[4];   // atom_mask      [N]
  const float* bmask = (const float*)d_in[5];   // batch_mask     [E]
  const float* acoef = (const float*)d_in[6];   // a_coefficient  [1]
  const float* aexp  = (const float*)d_in[7];   // a_exponent     [1]
  const float* phic  = (const float*)d_in[8];   // phi_coefficients [4]
  const float* phie  = (const float*)d_in[9];   // phi_exponents    [4]
  const int*   idx_i = (const int*)d_in[10];    // [E]
  const int*   idx_j = (const int*)d_in[11];    // [E]
  // d_in[12] batch_segments, d_in[13] batch_size: unused by the reference.

  const int N = in_sizes[0];
  const int E = in_sizes[1];

  float2* table = (float2*)d_ws;   // 1.6 MB: (za, safe_z) per atom
  float*  accum = (float*)d_out;   // accumulate segment_sum directly in d_out

  zbl_init_kernel<<<(N + 255) / 256, 256, 0, stream>>>(Z, aexp, table, accum, N);

  int tiles = (E + TILE - 1) / TILE;
  zbl_edges_kernel<<<tiles, BLOCK, 0, stream>>>(dist, swo, esh, bmask,
                                                idx_i, idx_j, table,
                                                acoef, phic, phie, accum, E);

  zbl_final_kernel<<<(N + 255) / 256, 256, 0, stream>>>(accum, amask, N);
}